// MVCG_3315714752652
// MI455X (gfx1250) — compile-verified
//
#include <hip/hip_runtime.h>
#include <hip/hip_bf16.h>
#include <math.h>

// Problem constants (from reference)
#define G_  4
#define N_  100000
#define E_  1600000
#define D_  64
#define B_  32
#define L_  50
#define CL_ 2

typedef __attribute__((ext_vector_type(2))) float v2f;
typedef __attribute__((ext_vector_type(8))) float v8f;

// ---------------------------------------------------------------- zero
__global__ void mvcg_zero(float* __restrict__ p, long n4) {
    long i = (long)blockIdx.x * blockDim.x + threadIdx.x;
    if (i < n4) {
        float4 z = make_float4(0.f, 0.f, 0.f, 0.f);
        *(float4*)(p + i * 4) = z;
    }
}

// ---------------------------------------------------------------- degree
__global__ void mvcg_degree(const int* __restrict__ dst, float* __restrict__ deg) {
    int g = blockIdx.y;
    int e = blockIdx.x * blockDim.x + threadIdx.x;
    if (e < E_) {
        int d = dst[(size_t)g * E_ + e];
        atomicAdd(&deg[(size_t)g * N_ + d], 1.0f);
    }
}

// ---------------------------------------------------------------- edge scatter
// 16 lanes per edge, float4 per lane -> 64 floats per edge into acc[dst]
__global__ void mvcg_scatter(const float* __restrict__ feat,
                             const int* __restrict__ src,
                             const int* __restrict__ dst,
                             const float* __restrict__ ew,
                             float* __restrict__ acc) {
    int g = blockIdx.y;
    long t = (long)blockIdx.x * blockDim.x + threadIdx.x;
    long e = t >> 4;
    int  q = (int)(t & 15);
    if (e >= E_) return;
    int   s = src[(size_t)g * E_ + e];
    int   d = dst[(size_t)g * E_ + e];
    float w = ew [(size_t)g * E_ + e];
    float4 v = *(const float4*)(feat + ((size_t)g * N_ + s) * D_ + q * 4);
    float* dp = acc + ((size_t)g * N_ + d) * D_ + q * 4;
    atomicAdd(dp + 0, v.x * w);
    atomicAdd(dp + 1, v.y * w);
    atomicAdd(dp + 2, v.z * w);
    atomicAdd(dp + 3, v.w * w);
}

// ---------------------------------------------------------------- SAGE GEMM
// out[n, :] = fin[n,:] @ Ws^T + (acc[n,:]/max(deg,1)) @ Wn^T + bias  (opt relu)
// One wave -> one 16x16 output tile via V_WMMA_F32_16X16X4_F32 (K=64 -> 16 steps x 2 inputs).
// Block = 256 threads = 8 waves covering 32 rows x 64 cols. NOT safe for fout==fin
// (4 waves share each 16-row block: each reads all 64 cols while the others write theirs).
__global__ void mvcg_gemm(const float* __restrict__ fin,
                          const float* __restrict__ acc,
                          const float* __restrict__ deg,
                          const float* __restrict__ Ws,
                          const float* __restrict__ Wn,
                          const float* __restrict__ bias,
                          float* __restrict__ fout,
                          int relu) {
    const int g    = blockIdx.y;
    const int wave = threadIdx.x >> 5;
    const int lane = threadIdx.x & 31;
    const int row0 = blockIdx.x * 32 + (wave & 1) * 16;
    const int j0   = (wave >> 1) * 16;
    const int half = lane >> 4;     // 0: lanes 0-15, 1: lanes 16-31
    const int l15  = lane & 15;

    v8f c = {0.f, 0.f, 0.f, 0.f, 0.f, 0.f, 0.f, 0.f};

#if __has_builtin(__builtin_amdgcn_wmma_f32_16x16x4_f32)
    // A frag (16x4 f32): lane<16 holds row l15, k0/k0+1; lane>=16 holds row l15, k0+2/k0+3
    // B frag (4x16 f32): lane<16 holds B[k0][j0+l15],B[k0+1][..]; lane>=16 -> k0+2/k0+3.
    // B[k][j] = W[j][k]  (out = X * W^T), so both A and B loads are float2 at offset 4*s+2*half.
    const int    arow = row0 + l15;
    const size_t abase = ((size_t)g * N_ + arow) * D_;
    const float* Wsr = Ws + (size_t)g * D_ * D_ + (size_t)(j0 + l15) * D_;
    const float* Wnr = Wn + (size_t)g * D_ * D_ + (size_t)(j0 + l15) * D_;

#pragma unroll
    for (int s = 0; s < 16; ++s) {
        const int k = 4 * s + 2 * half;
        v2f a = *(const v2f*)(fin + abase + k);
        v2f b = *(const v2f*)(Wsr + k);
        c = __builtin_amdgcn_wmma_f32_16x16x4_f32(false, a, false, b, (short)0, c,
                                                  false, false);
    }
    const float inv = 1.0f / fmaxf(deg[(size_t)g * N_ + arow], 1.0f);
#pragma unroll
    for (int s = 0; s < 16; ++s) {
        const int k = 4 * s + 2 * half;
        v2f a = *(const v2f*)(acc + abase + k);
        a.x *= inv; a.y *= inv;
        v2f b = *(const v2f*)(Wnr + k);
        c = __builtin_amdgcn_wmma_f32_16x16x4_f32(false, a, false, b, (short)0, c,
                                                  false, false);
    }
#else
    // Scalar fallback with identical C-fragment ownership (row0+r+8*half, j0+l15)
    for (int r = 0; r < 8; ++r) {
        const int row = row0 + r + 8 * half;
        const float* fr = fin + ((size_t)g * N_ + row) * D_;
        const float* ar = acc + ((size_t)g * N_ + row) * D_;
        const float* ws = Ws + (size_t)g * D_ * D_ + (size_t)(j0 + l15) * D_;
        const float* wn = Wn + (size_t)g * D_ * D_ + (size_t)(j0 + l15) * D_;
        const float invr = 1.0f / fmaxf(deg[(size_t)g * N_ + row], 1.0f);
        float sum = 0.f;
        for (int k = 0; k < D_; ++k)
            sum = fmaf(fr[k], ws[k], fmaf(ar[k] * invr, wn[k], sum));
        c[r] = sum;
    }
#endif

    // C/D layout: VGPR r -> lanes 0-15: (M=r, N=lane); lanes 16-31: (M=r+8, N=lane-16)
    const float bv = bias[(size_t)g * D_ + j0 + l15];
#pragma unroll
    for (int r = 0; r < 8; ++r) {
        const int row = row0 + r + 8 * half;
        float v = c[r] + bv;
        if (relu) v = fmaxf(v, 0.0f);
        fout[((size_t)g * N_ + row) * D_ + j0 + l15] = v;
    }
}

// ---------------------------------------------------------------- candidate gather
__global__ void mvcg_gather(const float* __restrict__ feat,
                            const int* __restrict__ cidx,
                            float* __restrict__ sel) {
    long t = (long)blockIdx.x * blockDim.x + threadIdx.x;
    if (t >= (long)B_ * G_ * L_ * D_) return;
    int  d  = (int)(t & 63);
    long r  = t >> 6;               // (b,g,l)
    int  l  = (int)(r % L_);
    long r2 = r / L_;
    int  g  = (int)(r2 % G_);
    int  b  = (int)(r2 / G_);
    int node = cidx[((size_t)b * G_ + g) * L_ + l];
    sel[t] = feat[((size_t)g * N_ + node) * D_ + d];
}

// ---------------------------------------------------------------- attention stage 1
// One block per (b,g): x is 50x64, two attention layers, sum over L -> y[b,g,:]
__global__ void mvcg_attn1(const float* __restrict__ sel,
                           const float* __restrict__ Q,
                           const float* __restrict__ Kw,
                           const float* __restrict__ Vw,
                           float* __restrict__ y) {
    __shared__ float x [L_][D_];
    __shared__ float A [L_][D_];   // q, later v
    __shared__ float Bm[L_][D_];   // k
    __shared__ float s [L_][L_];
    const int tid = threadIdx.x;
    const int bg = blockIdx.x;
    const int b = bg / G_, g = bg % G_;
    const float* xin = sel + ((size_t)b * G_ + g) * L_ * D_;

    for (int i = tid; i < L_ * D_; i += blockDim.x) x[i / D_][i % D_] = xin[i];
    __syncthreads();

    for (int layer = 0; layer < CL_; ++layer) {
        const float* Wq = Q  + (size_t)layer * D_ * D_;
        const float* Wk = Kw + (size_t)layer * D_ * D_;
        const float* Wv = Vw + (size_t)layer * D_ * D_;
        // q -> A, k -> Bm
        for (int i = tid; i < L_ * D_; i += blockDim.x) {
            int r = i / D_, cc = i % D_;
            float aq = 0.f, ak = 0.f;
            for (int k = 0; k < D_; ++k) {
                float xv = x[r][k];
                aq = fmaf(xv, Wq[cc * D_ + k], aq);
                ak = fmaf(xv, Wk[cc * D_ + k], ak);
            }
            A[r][cc] = aq; Bm[r][cc] = ak;
        }
        __syncthreads();
        // scores
        for (int i = tid; i < L_ * L_; i += blockDim.x) {
            int r = i / L_, m = i % L_;
            float d = 0.f;
            for (int k = 0; k < D_; ++k) d = fmaf(A[r][k], Bm[m][k], d);
            s[r][m] = d * 0.125f;   // 1/sqrt(64)
        }
        __syncthreads();
        // softmax rows
        if (tid < L_) {
            float mx = -1e30f;
            for (int m = 0; m < L_; ++m) mx = fmaxf(mx, s[tid][m]);
            float sum = 0.f;
            for (int m = 0; m < L_; ++m) { float e = expf(s[tid][m] - mx); s[tid][m] = e; sum += e; }
            float is = 1.0f / sum;
            for (int m = 0; m < L_; ++m) s[tid][m] *= is;
        }
        __syncthreads();
        // v -> A
        for (int i = tid; i < L_ * D_; i += blockDim.x) {
            int r = i / D_, cc = i % D_;
            float av = 0.f;
            for (int k = 0; k < D_; ++k) av = fmaf(x[r][k], Wv[cc * D_ + k], av);
            A[r][cc] = av;
        }
        __syncthreads();
        // x = softmax(s) @ v   (reads only s & A -> safe to overwrite x)
        for (int i = tid; i < L_ * D_; i += blockDim.x) {
            int r = i / D_, cc = i % D_;
            float o = 0.f;
            for (int m = 0; m < L_; ++m) o = fmaf(s[r][m], A[m][cc], o);
            x[r][cc] = o;
        }
        __syncthreads();
    }
    if (tid < D_) {
        float sum = 0.f;
        for (int l = 0; l < L_; ++l) sum += x[l][tid];
        y[((size_t)b * G_ + g) * D_ + tid] = sum;
    }
}

// ---------------------------------------------------------------- attention stage 2
// One block per b: x is 4x64 (over graphs), two layers, sum over G -> out[b,:]
__global__ void mvcg_attn2(const float* __restrict__ yin,
                           const float* __restrict__ Q,
                           const float* __restrict__ Kw,
                           const float* __restrict__ Vw,
                           float* __restrict__ out) {
    __shared__ float x [G_][D_];
    __shared__ float A [G_][D_];
    __shared__ float Bm[G_][D_];
    __shared__ float s [G_][G_];
    const int tid = threadIdx.x;
    const int b = blockIdx.x;

    for (int i = tid; i < G_ * D_; i += blockDim.x)
        x[i / D_][i % D_] = yin[(size_t)b * G_ * D_ + i];
    __syncthreads();

    for (int layer = 0; layer < CL_; ++layer) {
        const float* Wq = Q  + (size_t)layer * D_ * D_;
        const float* Wk = Kw + (size_t)layer * D_ * D_;
        const float* Wv = Vw + (size_t)layer * D_ * D_;
        for (int i = tid; i < G_ * D_; i += blockDim.x) {
            int r = i / D_, cc = i % D_;
            float aq = 0.f, ak = 0.f;
            for (int k = 0; k < D_; ++k) {
                float xv = x[r][k];
                aq = fmaf(xv, Wq[cc * D_ + k], aq);
                ak = fmaf(xv, Wk[cc * D_ + k], ak);
            }
            A[r][cc] = aq; Bm[r][cc] = ak;
        }
        __syncthreads();
        if (tid < G_ * G_) {
            int r = tid / G_, m = tid % G_;
            float d = 0.f;
            for (int k = 0; k < D_; ++k) d = fmaf(A[r][k], Bm[m][k], d);
            s[r][m] = d * 0.125f;
        }
        __syncthreads();
        if (tid < G_) {
            float mx = -1e30f;
            for (int m = 0; m < G_; ++m) mx = fmaxf(mx, s[tid][m]);
            float sum = 0.f;
            for (int m = 0; m < G_; ++m) { float e = expf(s[tid][m] - mx); s[tid][m] = e; sum += e; }
            float is = 1.0f / sum;
            for (int m = 0; m < G_; ++m) s[tid][m] *= is;
        }
        __syncthreads();
        for (int i = tid; i < G_ * D_; i += blockDim.x) {
            int r = i / D_, cc = i % D_;
            float av = 0.f;
            for (int k = 0; k < D_; ++k) av = fmaf(x[r][k], Wv[cc * D_ + k], av);
            A[r][cc] = av;
        }
        __syncthreads();
        for (int i = tid; i < G_ * D_; i += blockDim.x) {
            int r = i / D_, cc = i % D_;
            float o = 0.f;
            for (int m = 0; m < G_; ++m) o = fmaf(s[r][m], A[m][cc], o);
            x[r][cc] = o;
        }
        __syncthreads();
    }
    if (tid < D_) {
        float sum = 0.f;
        for (int r = 0; r < G_; ++r) sum += x[r][tid];
        out[(size_t)b * D_ + tid] = sum;
    }
}

// ---------------------------------------------------------------- launch
extern "C" void kernel_launch(void* const* d_in, const int* in_sizes, int n_in,
                              void* d_out, int out_size, void* d_ws, size_t ws_size,
                              hipStream_t stream) {
    const int*   src  = (const int*)  d_in[0];
    const int*   dst  = (const int*)  d_in[1];
    const float* ew   = (const float*)d_in[2];
    const float* emb  = (const float*)d_in[3];
    const float* Ws1  = (const float*)d_in[4];
    const float* Wn1  = (const float*)d_in[5];
    const float* b1   = (const float*)d_in[6];
    const float* Ws2  = (const float*)d_in[7];
    const float* Wn2  = (const float*)d_in[8];
    const float* b2   = (const float*)d_in[9];
    const float* Q1   = (const float*)d_in[10];
    const float* K1   = (const float*)d_in[11];
    const float* V1   = (const float*)d_in[12];
    const float* Q2   = (const float*)d_in[13];
    const float* K2   = (const float*)d_in[14];
    const float* V2   = (const float*)d_in[15];
    const int*   cidx = (const int*)  d_in[16];
    float* out = (float*)d_out;

    // workspace layout (floats): deg | acc | feat ; sel/y reuse acc after GNN
    float* ws   = (float*)d_ws;
    float* deg  = ws;
    float* acc  = ws + (size_t)G_ * N_;
    float* feat = acc + (size_t)G_ * N_ * D_;
    float* sel  = acc;                       // B*G*L*D = 409600 floats
    float* ybuf = acc + (1 << 20);           // B*G*D = 8192 floats, disjoint from sel

    const long nz_all = (long)G_ * N_ + (long)G_ * N_ * D_;  // deg + acc (contiguous)
    const long nz_acc = (long)G_ * N_ * D_;

    // 1) zero deg + acc
    mvcg_zero<<<dim3((unsigned)((nz_all / 4 + 255) / 256)), 256, 0, stream>>>(deg, nz_all / 4);
    // 2) degrees
    mvcg_degree<<<dim3((E_ + 255) / 256, G_), 256, 0, stream>>>(dst, deg);
    // 3) layer-1 scatter: acc += emb[src]*ew
    mvcg_scatter<<<dim3((unsigned)(((long)E_ * 16) / 256), G_), 256, 0, stream>>>(emb, src, dst, ew, acc);
    // 4) layer-1 GEMM (relu) -> feat
    mvcg_gemm<<<dim3(N_ / 32, G_), 256, 0, stream>>>(emb, acc, deg, Ws1, Wn1, b1, feat, 1);
    // 5) re-zero acc
    mvcg_zero<<<dim3((unsigned)((nz_acc / 4 + 255) / 256)), 256, 0, stream>>>(acc, nz_acc / 4);
    // 6) layer-2 scatter: acc += feat[src]*ew
    mvcg_scatter<<<dim3((unsigned)(((long)E_ * 16) / 256), G_), 256, 0, stream>>>(feat, src, dst, ew, acc);
    // 7) layer-2 GEMM (no relu), in place -> feat
    mvcg_gemm<<<dim3(N_ / 32, G_), 256, 0, stream>>>(feat, acc, deg, Ws2, Wn2, b2, feat, 0);
    // 8) gather candidates -> sel (reuses acc region; acc no longer needed)
    mvcg_gather<<<dim3((B_ * G_ * L_ * D_ + 255) / 256), 256, 0, stream>>>(feat, cidx, sel);
    // 9) attention over L=50, sum -> ybuf
    mvcg_attn1<<<dim3(B_ * G_), 256, 0, stream>>>(sel, Q1, K1, V1, ybuf);
    // 10) attention over G=4, sum -> out
    mvcg_attn2<<<dim3(B_), 64, 0, stream>>>(ybuf, Q2, K2, V2, out);
}